// Bfly_BertIntermediate_61349312856071
// MI455X (gfx1250) — compile-verified
//
#include <hip/hip_runtime.h>
#include <math.h>

typedef __attribute__((ext_vector_type(2))) float v2f;
typedef __attribute__((ext_vector_type(8))) float v8f;

#define TILE_ROWS 32
#define LDS_STRIDE 1028   // 1024 + 4 pad: B-frag column gathers hit distinct banks

// ws layout (float offsets)
#define OFF_A 0           // 4 stacks * 64 blocks * 256 floats (A-frag order)
#define OFF_B 65536       // 4 stacks * 64 (c,r)   * 256 floats (A-frag order)
#define OFF_C 131072      // 4 stacks * 256 q * 16 floats (row-major 4x4)

// ---------------- precompute: collapse butterfly layers into stage matrices --------------

// Stage A: layers 0..3 -> dense 16x16 per (stack, 16-block). One thread owns one column.
__global__ __launch_bounds__(256) void build_stageA(const float* __restrict__ tw,
                                                    float* __restrict__ matsA) {
    int tid = blockIdx.x * 256 + threadIdx.x;       // 4096 threads
    int ci = tid & 15;                              // column (input index K)
    int b  = (tid >> 4) & 63;                       // 16-block
    int k  = tid >> 10;                             // stack
    float v[16];
#pragma unroll
    for (int l = 0; l < 16; ++l) v[l] = (l == ci) ? 1.0f : 0.0f;
#pragma unroll
    for (int i = 0; i < 4; ++i) {
        int s = 1 << i;
#pragma unroll
        for (int l0 = 0; l0 < 16; ++l0) {
            if ((l0 >> i) & 1) continue;
            int l1 = l0 + s;
            int g0 = b * 16 + l0;
            int e  = (g0 >> (i + 1)) * s + (g0 & (s - 1));   // blk*stride + pos
            const float* t = tw + ((((k * 10 + i) << 9) + e) << 2);
            float x0 = v[l0], x1 = v[l1];
            v[l0] = t[0] * x0 + t[1] * x1;
            v[l1] = t[2] * x0 + t[3] * x1;
        }
    }
    // write in WMMA A-fragment order: elem(slice s, lane l, w) = M[l&15][4s + 2*(l>>4) + w]
    float* base = matsA + ((k * 64 + b) << 8);
    int sl = ci >> 2, hf = (ci >> 1) & 1, w = ci & 1;
#pragma unroll
    for (int m = 0; m < 16; ++m) base[sl * 64 + hf * 32 + 2 * m + w] = v[m];
}

// Stage B: layers 4..7 -> dense 16x16 over bit-field p (bits 4..7), per (stack, 256-block c, residue r).
__global__ __launch_bounds__(256) void build_stageB(const float* __restrict__ tw,
                                                    float* __restrict__ matsB) {
    int tid = blockIdx.x * 256 + threadIdx.x;       // 4096 threads
    int ci = tid & 15;                              // column (input p index)
    int r  = (tid >> 4) & 15;
    int c  = (tid >> 8) & 3;
    int k  = tid >> 10;
    float v[16];
#pragma unroll
    for (int l = 0; l < 16; ++l) v[l] = (l == ci) ? 1.0f : 0.0f;
#pragma unroll
    for (int i = 4; i < 8; ++i) {
        int s  = 1 << i;
        int ds = 1 << (i - 4);                      // stride in p-space
#pragma unroll
        for (int p0 = 0; p0 < 16; ++p0) {
            if ((p0 >> (i - 4)) & 1) continue;
            int p1 = p0 + ds;
            int g0 = c * 256 + p0 * 16 + r;
            int e  = (g0 >> (i + 1)) * s + (g0 & (s - 1));
            const float* t = tw + ((((k * 10 + i) << 9) + e) << 2);
            float x0 = v[p0], x1 = v[p1];
            v[p0] = t[0] * x0 + t[1] * x1;
            v[p1] = t[2] * x0 + t[3] * x1;
        }
    }
    float* base = matsB + ((k * 64 + c * 16 + r) << 8);
    int sl = ci >> 2, hf = (ci >> 1) & 1, w = ci & 1;
#pragma unroll
    for (int m = 0; m < 16; ++m) base[sl * 64 + hf * 32 + 2 * m + w] = v[m];
}

// Stage C: layers 8..9 -> 4x4 over bits 8..9, per (stack, low-8 residue q). Row-major store.
__global__ __launch_bounds__(256) void build_stageC(const float* __restrict__ tw,
                                                    float* __restrict__ matsC) {
    int tid = blockIdx.x * 256 + threadIdx.x;       // 1024 threads
    int q = tid & 255;
    int k = tid >> 8;
    float M[4][4];
#pragma unroll
    for (int z = 0; z < 4; ++z)
#pragma unroll
        for (int j = 0; j < 4; ++j) M[z][j] = (z == j) ? 1.0f : 0.0f;
    // layer 8 (stride 256): pairs (0,1),(2,3); e = (z0>>1)*256 + q
#pragma unroll
    for (int z0 = 0; z0 < 4; z0 += 2) {
        const float* t = tw + ((((k * 10 + 8) << 9) + ((z0 >> 1) << 8) + q) << 2);
#pragma unroll
        for (int j = 0; j < 4; ++j) {
            float x0 = M[z0][j], x1 = M[z0 + 1][j];
            M[z0][j]     = t[0] * x0 + t[1] * x1;
            M[z0 + 1][j] = t[2] * x0 + t[3] * x1;
        }
    }
    // layer 9 (stride 512): pairs (0,2),(1,3); e = z0*256 + q
#pragma unroll
    for (int z0 = 0; z0 < 2; ++z0) {
        const float* t = tw + ((((k * 10 + 9) << 9) + (z0 << 8) + q) << 2);
#pragma unroll
        for (int j = 0; j < 4; ++j) {
            float x0 = M[z0][j], x1 = M[z0 + 2][j];
            M[z0][j]     = t[0] * x0 + t[1] * x1;
            M[z0 + 2][j] = t[2] * x0 + t[3] * x1;
        }
    }
    float* base = matsC + (((k << 8) + q) << 4);
#pragma unroll
    for (int z = 0; z < 4; ++z)
#pragma unroll
        for (int j = 0; j < 4; ++j) base[z * 4 + j] = M[z][j];
}

// ---------------- main fused kernel: WMMA stages A,B + VALU stage C + bias + gelu ----------

__global__ __launch_bounds__(256) void bfly_main(const float* __restrict__ X,
                                                 const float* __restrict__ matsA,
                                                 const float* __restrict__ matsB,
                                                 const float* __restrict__ matsC,
                                                 const float* __restrict__ bias,
                                                 float* __restrict__ out) {
    __shared__ float xs[TILE_ROWS * LDS_STRIDE];    // 131,584 B working tile (in-place stages)
    __shared__ float cm[4096];                      // stage-C 4x4 mats for this stack
    __shared__ float bsh[1024];                     // bias slice for this stack

    const int tid   = threadIdx.x;
    const int stack = blockIdx.y;
    const int rowT  = blockIdx.x * TILE_ROWS;

    // preload stage-C matrices + bias (L2-resident)
    {
        const float4* src = (const float4*)(matsC + stack * 4096);
        float4* dst = (float4*)cm;
#pragma unroll
        for (int j = 0; j < 4; ++j) dst[j * 256 + tid] = src[j * 256 + tid];
        ((float4*)bsh)[tid] = ((const float4*)(bias + stack * 1024))[tid];
    }
    // load 32x1024 input tile (coalesced b128), padded LDS stride
#pragma unroll 4
    for (int it = 0; it < 32; ++it) {
        int flat = it * 256 + tid;                  // float4 index
        int row  = flat >> 8;
        int cg   = flat & 255;
        float4 vv = ((const float4*)(X + (size_t)(rowT + row) * 1024))[cg];
        *(float4*)(&xs[row * LDS_STRIDE + cg * 4]) = vv;
    }
    __syncthreads();

    const int wave = tid >> 5;
    const int lane = tid & 31;
    const int half = lane >> 4;
    const int nn   = lane & 15;

    // ---- Stage A: 128 block-GEMMs of 16x16x16 via 4 chained wmma_f32_16x16x4 ----
    for (int t = wave; t < 128; t += 8) {
        int st = t & 1, b = t >> 1;
        const float* Ab = matsA + (((stack << 6) + b) << 8);
        v2f a0 = *(const v2f*)(Ab +   0 + lane * 2);
        v2f a1 = *(const v2f*)(Ab +  64 + lane * 2);
        v2f a2 = *(const v2f*)(Ab + 128 + lane * 2);
        v2f a3 = *(const v2f*)(Ab + 192 + lane * 2);
        const float* xrow = &xs[(st * 16 + nn) * LDS_STRIDE + b * 16];
        v8f acc = {};
        v2f b0 = { xrow[2 * half + 0],  xrow[2 * half + 1]  };
        acc = __builtin_amdgcn_wmma_f32_16x16x4_f32(false, a0, false, b0, (short)0, acc, false, false);
        v2f b1 = { xrow[2 * half + 4],  xrow[2 * half + 5]  };
        acc = __builtin_amdgcn_wmma_f32_16x16x4_f32(false, a1, false, b1, (short)0, acc, false, false);
        v2f b2 = { xrow[2 * half + 8],  xrow[2 * half + 9]  };
        acc = __builtin_amdgcn_wmma_f32_16x16x4_f32(false, a2, false, b2, (short)0, acc, false, false);
        v2f b3 = { xrow[2 * half + 12], xrow[2 * half + 13] };
        acc = __builtin_amdgcn_wmma_f32_16x16x4_f32(false, a3, false, b3, (short)0, acc, false, false);
        float* orow = &xs[(st * 16 + nn) * LDS_STRIDE + b * 16 + 8 * half];
#pragma unroll
        for (int r = 0; r < 8; ++r) orow[r] = acc[r];
    }
    __syncthreads();

    // ---- Stage B: same, strided columns 256c + 16p + r ----
    for (int t = wave; t < 128; t += 8) {
        int st = t & 1, rem = t >> 1;
        int c = rem >> 4, r = rem & 15;
        const float* Ab = matsB + (((stack << 6) + rem) << 8);
        v2f a0 = *(const v2f*)(Ab +   0 + lane * 2);
        v2f a1 = *(const v2f*)(Ab +  64 + lane * 2);
        v2f a2 = *(const v2f*)(Ab + 128 + lane * 2);
        v2f a3 = *(const v2f*)(Ab + 192 + lane * 2);
        const float* xrow = &xs[(st * 16 + nn) * LDS_STRIDE + c * 256 + r];
        v8f acc = {};
        int p0 = 2 * half;
        v2f b0 = { xrow[(p0 + 0)  * 16], xrow[(p0 + 1)  * 16] };
        acc = __builtin_amdgcn_wmma_f32_16x16x4_f32(false, a0, false, b0, (short)0, acc, false, false);
        v2f b1 = { xrow[(p0 + 4)  * 16], xrow[(p0 + 5)  * 16] };
        acc = __builtin_amdgcn_wmma_f32_16x16x4_f32(false, a1, false, b1, (short)0, acc, false, false);
        v2f b2 = { xrow[(p0 + 8)  * 16], xrow[(p0 + 9)  * 16] };
        acc = __builtin_amdgcn_wmma_f32_16x16x4_f32(false, a2, false, b2, (short)0, acc, false, false);
        v2f b3 = { xrow[(p0 + 12) * 16], xrow[(p0 + 13) * 16] };
        acc = __builtin_amdgcn_wmma_f32_16x16x4_f32(false, a3, false, b3, (short)0, acc, false, false);
        float* orow = &xs[(st * 16 + nn) * LDS_STRIDE + c * 256 + r + (8 * half) * 16];
#pragma unroll
        for (int rr = 0; rr < 8; ++rr) orow[rr * 16] = acc[rr];
    }
    __syncthreads();

    // ---- Stage C (4x4 over bits 8..9) + bias + exact gelu + coalesced store ----
#pragma unroll 2
    for (int it = 0; it < 32; ++it) {
        int flat = it * 256 + tid;
        int row  = flat >> 8;
        int cg   = flat & 255;
        int colBase = cg * 4;
        int z = colBase >> 8;                        // constant within the float4 group
        const float* xr = &xs[row * LDS_STRIDE];
        float4 res;
        float* rp = (float*)&res;
#pragma unroll
        for (int i2 = 0; i2 < 4; ++i2) {
            int col = colBase + i2;
            int q   = col & 255;
            const float* cz = &cm[q * 16 + z * 4];
            float y = cz[0] * xr[q] + cz[1] * xr[256 + q] +
                      cz[2] * xr[512 + q] + cz[3] * xr[768 + q];
            y += bsh[col];
            rp[i2] = 0.5f * y * (1.0f + erff(y * 0.70710678118654752440f));
        }
        ((float4*)(out + (size_t)(rowT + row) * 4096 + stack * 1024))[cg] = res;
    }
}

// ---------------- launch ----------------

extern "C" void kernel_launch(void* const* d_in, const int* in_sizes, int n_in,
                              void* d_out, int out_size, void* d_ws, size_t ws_size,
                              hipStream_t stream) {
    const float* X    = (const float*)d_in[0];   // [4,4096,1024]
    const float* tw   = (const float*)d_in[1];   // [4,10,512,2,2]
    const float* bias = (const float*)d_in[2];   // [4096]
    float* out = (float*)d_out;                  // [4,4096,4096]
    float* ws  = (float*)d_ws;                   // needs 589,824 B

    float* matsA = ws + OFF_A;
    float* matsB = ws + OFF_B;
    float* matsC = ws + OFF_C;

    build_stageA<<<16, 256, 0, stream>>>(tw, matsA);
    build_stageB<<<16, 256, 0, stream>>>(tw, matsB);
    build_stageC<<<4, 256, 0, stream>>>(tw, matsC);
    bfly_main<<<dim3(512, 4), 256, 0, stream>>>(X, matsA, matsB, matsC, bias, out);
}